// ContourChamferLoss_31980326486171
// MI455X (gfx1250) — compile-verified
//
#include <hip/hip_runtime.h>
#include <hip/hip_bf16.h>

typedef float v2f __attribute__((ext_vector_type(2)));
typedef float v8f __attribute__((ext_vector_type(8)));

__device__ __forceinline__ unsigned umin32(unsigned a, unsigned b) {
    return a < b ? a : b;
}

// XOR-butterfly min across lanes using DS_SWIZZLE_B32 (group-of-32 mode:
// offset[15]=0, and_mask=0x1f, or_mask=0, xor_mask=MASK).
template <int MASK>
__device__ __forceinline__ void xor_min8(unsigned m[8]) {
    #pragma unroll
    for (int k = 0; k < 8; ++k) {
        unsigned o = (unsigned)__builtin_amdgcn_ds_swizzle(
            (int)m[k], (MASK << 10) | 0x1f);
        m[k] = umin32(m[k], o);
    }
}

// ---------------------------------------------------------------------------
// Kernel 0: zero the scalar output (d_out is poisoned by the harness).
// ---------------------------------------------------------------------------
__global__ void chamfer_zero_out(float* out) {
    if (threadIdx.x == 0 && blockIdx.x == 0) out[0] = 0.0f;
}

// ---------------------------------------------------------------------------
// Kernel 1: for each "owner" point (16 per block), min_j d^2(own, str_j) via
// V_WMMA_F32_16X16X4_F32 with the distance embedding:
//   A row i = [ px, py, |p|^2, 1 ]       (16x4, loop-invariant in VGPRs)
//   B col j = [ -2qx, -2qy, 1, |q|^2 ]   (4x16, streamed)
// => D = A x B (+0) is exactly the 16x16 tile of squared distances.
// Running min kept as uint bit patterns (valid ordering for d^2 >= 0),
// avoiding fminf's NaN-canonicalization VALU overhead.
// ---------------------------------------------------------------------------
__launch_bounds__(256)
__global__ void chamfer_tilemin(const float2* __restrict__ own,
                                const float2* __restrict__ str,
                                int str_tiles,
                                float* __restrict__ out_min)
{
    const int tid  = threadIdx.x;
    const int lane = tid & 31;
    const int wave = __builtin_amdgcn_readfirstlane(tid >> 5);  // uniform SGPR
    const int half = lane & 15;
    const bool hi  = lane >= 16;

    // Loop-invariant A tile (16 owner points).
    float2 pp = own[(size_t)blockIdx.x * 16 + half];
    v2f a;
    a.x = hi ? (pp.x * pp.x + pp.y * pp.y) : pp.x;   // K=0 / K=2
    a.y = hi ? 1.0f                        : pp.y;   // K=1 / K=3

    unsigned m[8];
    #pragma unroll
    for (int k = 0; k < 8; ++k) m[k] = 0x7F7FFFFFu;  // +FLT_MAX bit pattern

    // Uniform trip count per wave => scalar loop, EXEC stays all-ones (WMMA
    // requirement).  8 waves stride the stream tiles.
    const int iters = str_tiles >> 3;
    const float2* qp = str + (size_t)wave * 16 + half;

    #pragma unroll 4
    for (int it = 0; it < iters; ++it) {
        float2 qq = qp[(size_t)it * 128];  // 8 tiles * 16 points per step
        v2f b;
        b.x = hi ? 1.0f                        : (-2.0f * qq.x);  // K=0 / K=2
        b.y = hi ? (qq.x * qq.x + qq.y * qq.y) : (-2.0f * qq.y);  // K=1 / K=3

        v8f c = {};
        v8f d = __builtin_amdgcn_wmma_f32_16x16x4_f32(
            false, a, false, b, (short)0, c, false, false);

        #pragma unroll
        for (int k = 0; k < 8; ++k)
            m[k] = umin32(m[k], __float_as_uint(d[k]));
    }

    // Epilogue for str_tiles not divisible by 8 (wave-uniform predicate, so
    // participating waves still run WMMA with full EXEC).
    const int rem = str_tiles & 7;
    if (wave < rem) {
        float2 qq = str[(size_t)(iters * 8 + wave) * 16 + half];
        v2f b;
        b.x = hi ? 1.0f                        : (-2.0f * qq.x);
        b.y = hi ? (qq.x * qq.x + qq.y * qq.y) : (-2.0f * qq.y);
        v8f c = {};
        v8f d = __builtin_amdgcn_wmma_f32_16x16x4_f32(
            false, a, false, b, (short)0, c, false, false);
        #pragma unroll
        for (int k = 0; k < 8; ++k)
            m[k] = umin32(m[k], __float_as_uint(d[k]));
    }

    // Row-min butterfly over each 16-lane half (halves never mix: lanes 0-15
    // carry rows 0..7, lanes 16-31 carry rows 8..15 of the D layout).
    xor_min8<1>(m);
    xor_min8<2>(m);
    xor_min8<4>(m);
    xor_min8<8>(m);

    // Combine the 8 waves through LDS.
    __shared__ unsigned red[8][16];
    if (lane == 0) {
        #pragma unroll
        for (int k = 0; k < 8; ++k) red[wave][k] = m[k];
    } else if (lane == 16) {
        #pragma unroll
        for (int k = 0; k < 8; ++k) red[wave][8 + k] = m[k];
    }
    __syncthreads();

    if (tid < 16) {
        unsigned v = red[0][tid];
        #pragma unroll
        for (int w = 1; w < 8; ++w) v = umin32(v, red[w][tid]);
        out_min[(size_t)blockIdx.x * 16 + tid] = __uint_as_float(v);
    }
}

// ---------------------------------------------------------------------------
// Kernel 2: chamfer = sum over all per-point min-d^2 of sqrt(max(.,0)).
// ---------------------------------------------------------------------------
__launch_bounds__(256)
__global__ void chamfer_sumsqrt(const float* __restrict__ mins, int n,
                                float* __restrict__ out)
{
    __shared__ float s[256];
    float acc = 0.0f;
    for (int i = blockIdx.x * blockDim.x + threadIdx.x; i < n;
         i += gridDim.x * blockDim.x)
        acc += sqrtf(fmaxf(mins[i], 0.0f));
    s[threadIdx.x] = acc;
    __syncthreads();
    #pragma unroll
    for (int off = 128; off > 0; off >>= 1) {
        if (threadIdx.x < off) s[threadIdx.x] += s[threadIdx.x + off];
        __syncthreads();
    }
    if (threadIdx.x == 0) atomicAdd(out, s[0]);
}

// ---------------------------------------------------------------------------
extern "C" void kernel_launch(void* const* d_in, const int* in_sizes, int n_in,
                              void* d_out, int out_size, void* d_ws, size_t ws_size,
                              hipStream_t stream)
{
    const float2* p = (const float2*)d_in[0];   // img_render_points: 16384 x (x,y)
    const float2* q = (const float2*)d_in[1];   // contour_points:    16384 x (x,y)
    const int Np = in_sizes[0] / 2;             // 16384
    const int Nq = in_sizes[1] / 2;             // 16384

    float* rowmin = (float*)d_ws;               // Np floats
    float* colmin = rowmin + Np;                // Nq floats (contiguous)
    float* out    = (float*)d_out;

    chamfer_zero_out<<<1, 64, 0, stream>>>(out);

    // Pass 1: per-render-point min over contour points.
    chamfer_tilemin<<<Np / 16, 256, 0, stream>>>(p, q, Nq / 16, rowmin);
    // Pass 2 (d^2 symmetric): per-contour-point min over render points.
    chamfer_tilemin<<<Nq / 16, 256, 0, stream>>>(q, p, Np / 16, colmin);

    chamfer_sumsqrt<<<64, 256, 0, stream>>>(rowmin, Np + Nq, out);
}